// clique_MPNN_7481833029838
// MI455X (gfx1250) — compile-verified
//
#include <hip/hip_runtime.h>

#define NNODES 32768
#define NGRAPH 128
#define PPG    256
#define DFEAT  512
#define H1DIM  64
#define EPG    2048          // edges per graph (P*DEG)
#define NEGS   0.01f

typedef __attribute__((ext_vector_type(2))) float v2f;
typedef __attribute__((ext_vector_type(8))) float v8f;

__device__ __forceinline__ float actf(float v, float slope) {
  return v > 0.0f ? v : v * slope;   // slope=0 -> relu, slope=1 -> identity
}

// ---------------------------------------------------------------------------
// FP32 WMMA GEMM:  out[N x M] = act(A[N x K] @ B[K x M] + bias)
// One wave computes a 32x64 tile (2 row-tiles x 4 col-frags = 8 accumulators),
// so each B fragment feeds two WMMAs: 8 wmma per 12 loads per K-step.
// 8 waves/block share the same 32 A-rows (waves = column groups for M=512).
// A-frag layout (ISA 16x4 f32): lane holds A[row + lane%16, k0 + 2*(lane/16) + j]
// B-frag layout (mirrored):     lane holds B[k0 + 2*(lane/16) + j, col + lane%16]
// C/D layout: vgpr r, lane l -> D[r + 8*(l/16), l%16]
// ---------------------------------------------------------------------------
template <int K, int M>
__global__ __launch_bounds__(256) void gemm_bias_act(const float* __restrict__ A,
                                                     const float* __restrict__ B,
                                                     const float* __restrict__ bias,
                                                     float* __restrict__ out,
                                                     float slope) {
  constexpr int NCG = M / 64;                 // column groups of 64
  const int lane = threadIdx.x & 31;
  const int wave = threadIdx.x >> 5;
  const int tile = blockIdx.x * 8 + wave;
  const int row  = (tile / NCG) * 32;
  const int colB = (tile % NCG) * 64;
  const int rl   = lane & 15;
  const int kh   = (lane >> 4) * 2;           // 0 or 2

  v8f c00 = {}; v8f c01 = {}; v8f c02 = {}; v8f c03 = {};
  v8f c10 = {}; v8f c11 = {}; v8f c12 = {}; v8f c13 = {};
  const float* aRow0 = A + (size_t)(row + rl) * K + kh;
  const float* aRow1 = A + (size_t)(row + 16 + rl) * K + kh;

  for (int k0 = 0; k0 < K; k0 += 4) {
    v2f a0 = *(const v2f*)(aRow0 + k0);
    v2f a1 = *(const v2f*)(aRow1 + k0);
    const float* bp = B + (size_t)(k0 + kh) * M + colB + rl;
    v2f b0 = { bp[0],  bp[M + 0]  };
    v2f b1 = { bp[16], bp[M + 16] };
    v2f b2 = { bp[32], bp[M + 32] };
    v2f b3 = { bp[48], bp[M + 48] };
    c00 = __builtin_amdgcn_wmma_f32_16x16x4_f32(false, a0, false, b0, (short)0, c00, false, false);
    c01 = __builtin_amdgcn_wmma_f32_16x16x4_f32(false, a0, false, b1, (short)0, c01, false, false);
    c02 = __builtin_amdgcn_wmma_f32_16x16x4_f32(false, a0, false, b2, (short)0, c02, false, false);
    c03 = __builtin_amdgcn_wmma_f32_16x16x4_f32(false, a0, false, b3, (short)0, c03, false, false);
    c10 = __builtin_amdgcn_wmma_f32_16x16x4_f32(false, a1, false, b0, (short)0, c10, false, false);
    c11 = __builtin_amdgcn_wmma_f32_16x16x4_f32(false, a1, false, b1, (short)0, c11, false, false);
    c12 = __builtin_amdgcn_wmma_f32_16x16x4_f32(false, a1, false, b2, (short)0, c12, false, false);
    c13 = __builtin_amdgcn_wmma_f32_16x16x4_f32(false, a1, false, b3, (short)0, c13, false, false);
  }

  const float bb0 = bias[colB + rl +  0];
  const float bb1 = bias[colB + rl + 16];
  const float bb2 = bias[colB + rl + 32];
  const float bb3 = bias[colB + rl + 48];
  const int mh = (lane >> 4) * 8;
#pragma unroll
  for (int r = 0; r < 8; ++r) {
    float* op0 = out + (size_t)(row + r + mh) * M + colB + rl;
    op0[0]  = actf(c00[r] + bb0, slope);
    op0[16] = actf(c01[r] + bb1, slope);
    op0[32] = actf(c02[r] + bb2, slope);
    op0[48] = actf(c03[r] + bb3, slope);
    float* op1 = out + (size_t)(row + 16 + r + mh) * M + colB + rl;
    op1[0]  = actf(c10[r] + bb0, slope);
    op1[16] = actf(c11[r] + bb1, slope);
    op1[32] = actf(c12[r] + bb2, slope);
    op1[48] = actf(c13[r] + bb3, slope);
  }
}

// ---------------------------------------------------------------------------
// Small helper kernels
// ---------------------------------------------------------------------------
__global__ void count_nodes(const int* __restrict__ batch, float* __restrict__ cnt) {
  int n = blockIdx.x * blockDim.x + threadIdx.x;
  if (n < NNODES) atomicAdd(&cnt[batch[n]], 1.0f);
}

__global__ void scatter_add_scalar(const float* __restrict__ v, const int* __restrict__ src,
                                   const int* __restrict__ dst, float* __restrict__ out, int E) {
  int e = blockIdx.x * blockDim.x + threadIdx.x;
  if (e < E) atomicAdd(&out[dst[e]], v[src[e]]);
}

__global__ void scatter_add_vec(const float* __restrict__ h, const int* __restrict__ src,
                                const int* __restrict__ dst, float* __restrict__ out) {
  int e = blockIdx.x;
  const float* ps = h  + (size_t)src[e] * DFEAT;
  float*       pd = out + (size_t)dst[e] * DFEAT;
  for (int f = threadIdx.x; f < DFEAT; f += blockDim.x) atomicAdd(&pd[f], ps[f]);
}

__global__ void binarize(const float* __restrict__ in, float* __restrict__ out) {
  int n = blockIdx.x * blockDim.x + threadIdx.x;
  if (n < NNODES) out[n] = in[n] > 0.0f ? 1.0f : 0.0f;
}

__global__ void scale_by_eps(const float* __restrict__ in, const float* __restrict__ eps,
                             float* __restrict__ out, size_t ncount) {
  size_t i = (size_t)blockIdx.x * blockDim.x + threadIdx.x;
  if (i < ncount) out[i] = (1.0f + eps[0]) * in[i];
}

__global__ void outer_relu(const float* __restrict__ h2s, const float* __restrict__ W1,
                           const float* __restrict__ b1, float* __restrict__ out) {
  int n = blockIdx.x;
  float hv = h2s[n];
  for (int d = threadIdx.x; d < DFEAT; d += blockDim.x)
    out[(size_t)n * DFEAT + d] = fmaxf(hv * W1[d] + b1[d], 0.0f);
}

// per-feature sum / sumsq over N rows; grid (2, 64), block 256
__global__ void col_stats(const float* __restrict__ in, float* __restrict__ stats) {
  int f  = blockIdx.x * 256 + threadIdx.x;
  int r0 = blockIdx.y * (NNODES / 64);
  float s = 0.0f, ss = 0.0f;
  for (int r = r0; r < r0 + NNODES / 64; ++r) {
    float v = in[(size_t)r * DFEAT + f];
    s += v; ss += v * v;
  }
  atomicAdd(&stats[f], s);
  atomicAdd(&stats[DFEAT + f], ss);
}

// out = (residual? residual + : ) act(g*(in-m)*rsqrt(v+1e-5)+b, slope)
__global__ void bn_apply(const float* __restrict__ in, const float* __restrict__ stats,
                         const float* __restrict__ g, const float* __restrict__ b,
                         const float* __restrict__ residual, float* __restrict__ out,
                         float slope) {
  size_t i = (size_t)blockIdx.x * blockDim.x + threadIdx.x;
  int d = (int)(i & (DFEAT - 1));
  float m  = stats[d] * (1.0f / NNODES);
  float vv = stats[DFEAT + d] * (1.0f / NNODES) - m * m;
  float v  = g[d] * (in[i] - m) * rsqrtf(vv + 1e-5f) + b[d];
  v = actf(v, slope);
  if (residual) v += residual[i];
  out[i] = v;
}

__global__ void make_rowscale(const float* __restrict__ mask, const float* __restrict__ cnt,
                              const int* __restrict__ batch, float* __restrict__ rs) {
  int n = blockIdx.x * blockDim.x + threadIdx.x;
  if (n < NNODES) rs[n] = mask[n] * rsqrtf(cnt[batch[n]]);
}

__global__ void row_scale(const float* __restrict__ in, const float* __restrict__ rs,
                          float* __restrict__ out) {
  size_t i = (size_t)blockIdx.x * blockDim.x + threadIdx.x;
  out[i] = in[i] * rs[i >> 9];   // D = 512
}

__global__ void lin2_mask(const float* __restrict__ h1, const float* __restrict__ W2,
                          const float* __restrict__ b2, const float* __restrict__ mask,
                          float* __restrict__ s1) {
  int n = blockIdx.x * blockDim.x + threadIdx.x;
  if (n >= NNODES) return;
  float acc = 0.0f;
  const float* hp = h1 + (size_t)n * H1DIM;
#pragma unroll
  for (int k = 0; k < H1DIM; ++k) acc += hp[k] * W2[k];
  float v = mask[n] * acc + b2[0];
  v = actf(v, NEGS);
  s1[n] = v * mask[n];
}

// block g reduces its 256 contiguous nodes: min/max -> probs -> sums
__global__ __launch_bounds__(PPG) void graph_reduce(const float* __restrict__ s1,
                                                    const float* __restrict__ noise,
                                                    float* __restrict__ probs,
                                                    float* __restrict__ psum,
                                                    float* __restrict__ sself,
                                                    float* __restrict__ setsize) {
  __shared__ float sd[PPG];
  int g = blockIdx.x, t = threadIdx.x;
  int n = g * PPG + t;
  float v = s1[n];

  sd[t] = v; __syncthreads();
  for (int s = PPG / 2; s > 0; s >>= 1) { if (t < s) sd[t] = fmaxf(sd[t], sd[t + s]); __syncthreads(); }
  float bmax = sd[0]; __syncthreads();

  sd[t] = v; __syncthreads();
  for (int s = PPG / 2; s > 0; s >>= 1) { if (t < s) sd[t] = fminf(sd[t], sd[t + s]); __syncthreads(); }
  float bmin = sd[0]; __syncthreads();

  float p = (v - bmin) / (bmax + 1e-6f - bmin);
  probs[n] = p;
  float x2 = (p - noise[n]) > 0.0f ? 1.0f : 0.0f;

  sd[t] = p; __syncthreads();
  for (int s = PPG / 2; s > 0; s >>= 1) { if (t < s) sd[t] += sd[t + s]; __syncthreads(); }
  if (t == 0) psum[g] = sd[0];
  __syncthreads();

  sd[t] = p * p; __syncthreads();
  for (int s = PPG / 2; s > 0; s >>= 1) { if (t < s) sd[t] += sd[t + s]; __syncthreads(); }
  if (t == 0) sself[g] = sd[0];
  __syncthreads();

  sd[t] = x2; __syncthreads();
  for (int s = PPG / 2; s > 0; s >>= 1) { if (t < s) sd[t] += sd[t + s]; __syncthreads(); }
  if (t == 0) setsize[g] = sd[0];
}

__global__ void edge_weight(const float* __restrict__ probs, const int* __restrict__ src,
                            const int* __restrict__ dst, float* __restrict__ ew) {
  __shared__ float sd[256];
  int g = blockIdx.x, t = threadIdx.x;
  float acc = 0.0f;
  for (int e = g * EPG + t; e < (g + 1) * EPG; e += 256) acc += probs[src[e]] * probs[dst[e]];
  sd[t] = acc; __syncthreads();
  for (int s = 128; s > 0; s >>= 1) { if (t < s) sd[t] += sd[t + s]; __syncthreads(); }
  if (t == 0) ew[g] = sd[0] * 0.5f;
}

__global__ void loss_kernel(const float* __restrict__ psum, const float* __restrict__ sself,
                            const float* __restrict__ ew, float* __restrict__ loss) {
  int g = threadIdx.x;
  if (g < NGRAPH) {
    float pairwise = psum[g] * psum[g] * 0.5f;
    float ecw = pairwise - sself[g];
    float ed  = ecw - ew[g];
    loss[g] = 0.25f * ed * 0.5f - 0.5f * ew[g];
  }
}

// ---------------------------------------------------------------------------
extern "C" void kernel_launch(void* const* d_in, const int* in_sizes, int n_in,
                              void* d_out, int out_size, void* d_ws, size_t ws_size,
                              hipStream_t stream) {
  const float* x      = (const float*)d_in[0];
  const float* noise  = (const float*)d_in[1];
  const int*   ei     = (const int*)d_in[2];
  const int*   batch  = (const int*)d_in[3];
  const float* c1_W1  = (const float*)d_in[4];
  const float* c1_b1  = (const float*)d_in[5];
  const float* c1_W2  = (const float*)d_in[6];
  const float* c1_b2  = (const float*)d_in[7];
  const float* c1_eps = (const float*)d_in[8];
  const float* c1_bng = (const float*)d_in[9];
  const float* c1_bnb = (const float*)d_in[10];
  const float* bn1_g  = (const float*)d_in[11];
  const float* bn1_b  = (const float*)d_in[12];
  const float* L_W1   = (const float*)d_in[13];
  const float* L_b1   = (const float*)d_in[14];
  const float* L_W2   = (const float*)d_in[15];
  const float* L_b2   = (const float*)d_in[16];
  const float* L_eps  = (const float*)d_in[17];
  const float* L_bng  = (const float*)d_in[18];
  const float* L_bnb  = (const float*)d_in[19];
  const float* Lo_g   = (const float*)d_in[20];
  const float* Lo_b   = (const float*)d_in[21];
  const float* l1_W   = (const float*)d_in[22];
  const float* l1_b   = (const float*)d_in[23];
  const float* l2_W   = (const float*)d_in[24];
  const float* l2_b   = (const float*)d_in[25];

  const int E = in_sizes[2] / 2;
  const int* src = ei;
  const int* dst = ei + E;

  const size_t ND = (size_t)NNODES * DFEAT;
  float* H    = (float*)d_ws;
  float* Abuf = H + ND;
  float* Bbuf = Abuf + ND;
  float* mask = Bbuf + ND;
  float* tmpN = mask + NNODES;
  float* h2s  = tmpN + NNODES;
  float* rs   = h2s + NNODES;
  float* s1   = rs + NNODES;
  float* stats= s1 + NNODES;          // 2*DFEAT
  float* cnt  = stats + 2 * DFEAT;    // NGRAPH
  float* psum = cnt + NGRAPH;
  float* sself= psum + NGRAPH;
  float* ew   = sself + NGRAPH;

  float* outLoss  = (float*)d_out;
  float* outProbs = outLoss + NGRAPH;
  float* outSize  = outProbs + NNODES;

  const int TB = 256;
  const int ndBlocks = (int)(ND / TB);                     // 65536
  const int nBlocks  = (NNODES + TB - 1) / TB;
  const int eBlocks  = (E + TB - 1) / TB;
  const int gemmBlks = (NNODES / 32) * (DFEAT / 64) / 8;   // 1024
  const int linBlks  = (NNODES / 32) * (H1DIM / 64) / 8;   // 128

  // per-graph node counts (for gnorm)
  (void)hipMemsetAsync(cnt, 0, NGRAPH * sizeof(float), stream);
  count_nodes<<<nBlocks, TB, 0, stream>>>(batch, cnt);

  // mask = 1hop(x)
  (void)hipMemcpyAsync(tmpN, x, NNODES * sizeof(float), hipMemcpyDeviceToDevice, stream);
  scatter_add_scalar<<<eBlocks, TB, 0, stream>>>(x, src, dst, tmpN, E);
  binarize<<<nBlocks, TB, 0, stream>>>(tmpN, mask);

  // --- c1 GIN: h2 = (1+eps)*x + agg(x); t1 = relu(h2*W1+b1); t2 = relu(t1@W2+b2)
  scale_by_eps<<<nBlocks, TB, 0, stream>>>(x, c1_eps, h2s, (size_t)NNODES);
  scatter_add_scalar<<<eBlocks, TB, 0, stream>>>(x, src, dst, h2s, E);
  outer_relu<<<NNODES, TB, 0, stream>>>(h2s, c1_W1, c1_b1, Abuf);
  gemm_bias_act<DFEAT, DFEAT><<<gemmBlks, TB, 0, stream>>>(Abuf, c1_W2, c1_b2, Bbuf, 0.0f);
  (void)hipMemsetAsync(stats, 0, 2 * DFEAT * sizeof(float), stream);
  col_stats<<<dim3(2, 64), TB, 0, stream>>>(Bbuf, stats);
  bn_apply<<<ndBlocks, TB, 0, stream>>>(Bbuf, stats, c1_bng, c1_bnb, (const float*)nullptr, H, NEGS);

  // h = bn1(gnorm(h*mask))
  make_rowscale<<<nBlocks, TB, 0, stream>>>(mask, cnt, batch, rs);
  row_scale<<<ndBlocks, TB, 0, stream>>>(H, rs, Abuf);
  (void)hipMemsetAsync(stats, 0, 2 * DFEAT * sizeof(float), stream);
  col_stats<<<dim3(2, 64), TB, 0, stream>>>(Abuf, stats);
  bn_apply<<<ndBlocks, TB, 0, stream>>>(Abuf, stats, bn1_g, bn1_b, (const float*)nullptr, H, 1.0f);

  // --- 3 GIN layers
  for (int i = 0; i < 3; ++i) {
    const float* W1  = L_W1 + (size_t)i * DFEAT * DFEAT;
    const float* b1  = L_b1 + (size_t)i * DFEAT;
    const float* W2  = L_W2 + (size_t)i * DFEAT * DFEAT;
    const float* b2  = L_b2 + (size_t)i * DFEAT;
    const float* bng = L_bng + (size_t)i * DFEAT;
    const float* bnb = L_bnb + (size_t)i * DFEAT;
    const float* og  = Lo_g + (size_t)i * DFEAT;
    const float* ob  = Lo_b + (size_t)i * DFEAT;

    // h2 = (1+eps)*h + agg(h)
    scale_by_eps<<<ndBlocks, TB, 0, stream>>>(H, L_eps + i, Abuf, ND);
    scatter_add_vec<<<E, TB, 0, stream>>>(H, src, dst, Abuf);
    // two GEMMs (fused bias+relu)
    gemm_bias_act<DFEAT, DFEAT><<<gemmBlks, TB, 0, stream>>>(Abuf, W1, b1, Bbuf, 0.0f);
    gemm_bias_act<DFEAT, DFEAT><<<gemmBlks, TB, 0, stream>>>(Bbuf, W2, b2, Abuf, 0.0f);
    // h = h + leaky(bn(t2))
    (void)hipMemsetAsync(stats, 0, 2 * DFEAT * sizeof(float), stream);
    col_stats<<<dim3(2, 64), TB, 0, stream>>>(Abuf, stats);
    bn_apply<<<ndBlocks, TB, 0, stream>>>(Abuf, stats, bng, bnb, H, H, NEGS);
    // mask = 1hop(mask)
    (void)hipMemcpyAsync(tmpN, mask, NNODES * sizeof(float), hipMemcpyDeviceToDevice, stream);
    scatter_add_scalar<<<eBlocks, TB, 0, stream>>>(mask, src, dst, tmpN, E);
    binarize<<<nBlocks, TB, 0, stream>>>(tmpN, mask);
    // h = bn_out(gnorm(h*mask))
    make_rowscale<<<nBlocks, TB, 0, stream>>>(mask, cnt, batch, rs);
    row_scale<<<ndBlocks, TB, 0, stream>>>(H, rs, Abuf);
    (void)hipMemsetAsync(stats, 0, 2 * DFEAT * sizeof(float), stream);
    col_stats<<<dim3(2, 64), TB, 0, stream>>>(Abuf, stats);
    bn_apply<<<ndBlocks, TB, 0, stream>>>(Abuf, stats, og, ob, (const float*)nullptr, H, 1.0f);
  }

  // --- readout
  gemm_bias_act<DFEAT, H1DIM><<<linBlks, TB, 0, stream>>>(H, l1_W, l1_b, Bbuf, NEGS);
  lin2_mask<<<nBlocks, TB, 0, stream>>>(Bbuf, l2_W, l2_b, mask, s1);
  graph_reduce<<<NGRAPH, PPG, 0, stream>>>(s1, noise, outProbs, psum, sself, outSize);
  edge_weight<<<NGRAPH, TB, 0, stream>>>(outProbs, src, dst, ew);
  loss_kernel<<<1, NGRAPH, 0, stream>>>(psum, sself, ew, outLoss);

  (void)n_in; (void)out_size; (void)ws_size;
}